// OT_Loss_72009421685332
// MI455X (gfx1250) — compile-verified
//
#include <hip/hip_runtime.h>
#include <hip/hip_bf16.h>
#include <math.h>

// Problem constants (match reference)
#define S_DIM   128
#define SS      16384
#define NPTS    512
#define NIMG    8
#define REG_F   10.0f
#define M_EPSF  1e-16f
#define N_ITER  100
#define THREADS 1024

// Workspace layout (floats)
#define KYT_OFF 0                    // per image: KyT[yy*512+n], 65536 floats
#define KX_OFF  65536                // per image: Kx [n*128+xx], 65536 floats
#define KXT_OFF 131072               // per image: KxT[xx*512+n], 65536 floats
#define IMG_WS  196608
#define PART_OFF (NIMG * IMG_WS)     // + img*4 + {loss, wd, ot}

typedef float v2f __attribute__((ext_vector_type(2)));
typedef float v8f __attribute__((ext_vector_type(8)));

__device__ __forceinline__ v8f wmma4(v2f a, v2f b, v8f c) {
  // D = A(16x4 f32) * B(4x16 f32) + C(16x16 f32)
  return __builtin_amdgcn_wmma_f32_16x16x4_f32(false, a, false, b, (short)0, c,
                                               false, false);
}

// Positive-denominator fast divide: v_rcp_f32 + 1 Newton step (~1 ulp).
__device__ __forceinline__ float fast_div(float n, float d) {
  float r = __builtin_amdgcn_rcpf(d);
  r = r * (2.0f - d * r);
  return n * r;
}

__device__ __forceinline__ float blockReduceSum(float val, float* red) {
  const int tid = threadIdx.x;
  red[tid] = val;
  __syncthreads();
  for (int off = THREADS / 2; off > 0; off >>= 1) {
    if (tid < off) red[tid] += red[tid + off];
    __syncthreads();
  }
  float r = red[0];
  __syncthreads();
  return r;
}

__global__ void __launch_bounds__(THREADS)
ot_sinkhorn_kernel(const float* __restrict__ nd,   // (8,1,128,128) normed density
                   const float* __restrict__ ud,   // (8,1,128,128) unnormed density
                   const float* __restrict__ pts,  // (8,512,2)
                   float* __restrict__ ws) {
  __shared__ float v_lds[SS];      // 64 KB: v vector as 128x128
  __shared__ float u_lds[NPTS];    // 2 KB
  __shared__ float red[THREADS];   // 4 KB reduction scratch

  const int img  = blockIdx.x;
  const int tid  = threadIdx.x;
  const int lane = tid & 31;
  const int wave = tid >> 5;       // 32 waves
  const int hi   = lane >> 4;      // 0/1: which K/M half this lane owns
  const int lm   = lane & 15;

  float* KyT = ws + (size_t)img * IMG_WS + KYT_OFF;  // [yy*512 + n]
  float* Kx  = ws + (size_t)img * IMG_WS + KX_OFF;   // [n*128 + xx]
  float* KxT = ws + (size_t)img * IMG_WS + KXT_OFF;  // [xx*512 + n]
  const float* b  = nd  + (size_t)img * SS;
  const float* sd = ud  + (size_t)img * SS;
  const float* P  = pts + (size_t)img * NPTS * 2;

  // ---------------- setup: separable kernel factors ----------------
  // dis[n,(yy,xx)] = dy[n,yy]+dx[n,xx];  K = exp(-dis/REG) = Ky*Kx
  for (int i = tid; i < NPTS * S_DIM; i += THREADS) {
    int n = i >> 7, xx = i & (S_DIM - 1);
    float x = P[n * 2 + 0];
    float c = 4.0f * (float)xx + 2.0f;
    float d = -2.0f * x * c + x * x + c * c;   // same association as reference
    Kx[n * S_DIM + xx] = expf(-d / REG_F);
  }
  for (int i = tid; i < S_DIM * NPTS; i += THREADS) {
    int xx = i >> 9, n = i & (NPTS - 1);
    float x = P[n * 2 + 0];
    float c = 4.0f * (float)xx + 2.0f;
    float d = -2.0f * x * c + x * x + c * c;
    KxT[xx * NPTS + n] = expf(-d / REG_F);     // bit-identical transpose copy
  }
  for (int i = tid; i < S_DIM * NPTS; i += THREADS) {
    int yy = i >> 9, n = i & (NPTS - 1);
    float y = P[n * 2 + 1];
    float c = 4.0f * (float)yy + 2.0f;
    float d = -2.0f * y * c + y * y + c * c;
    KyT[yy * NPTS + n] = expf(-d / REG_F);
  }
  if (tid < NPTS) u_lds[tid] = 1.0f / (float)NPTS;
  __threadfence();
  __syncthreads();

  // Phase-1 tile assignment: wave owns tiles (ty, tx0) and (ty, tx0+1),
  // which share the same A fragment (row strip of u*KyT).
  const int ty  = wave >> 2;        // 0..7
  const int tx0 = (wave & 3) * 2;   // 0,2,4,6

  // ---------------- Sinkhorn loop, fully on-chip ----------------
#pragma unroll 1
  for (int it = 0; it < N_ITER; ++it) {
    // Phase 1: R[yy,xx] = sum_n u[n]*Ky[n,yy]*Kx[n,xx]  (128x128x512 GEMM)
    //          v = b / (R + eps)
    {
      v8f acc0 = {0.f, 0.f, 0.f, 0.f, 0.f, 0.f, 0.f, 0.f};
      v8f acc1 = {0.f, 0.f, 0.f, 0.f, 0.f, 0.f, 0.f, 0.f};
      const float* kyrow = KyT + (ty * 16 + lm) * NPTS;        // A row m=lm
      const float* kxc0  = KxT + (tx0 * 16 + lm) * NPTS;       // B col n=lm
      const float* kxc1  = KxT + ((tx0 + 1) * 16 + lm) * NPTS; // (k-contiguous)
#pragma unroll 4
      for (int k0 = 0; k0 < NPTS; k0 += 4) {
        const int k = k0 + 2 * hi;                             // lane's K pair
        v2f a, b0, b1;
        a.x  = kyrow[k]     * u_lds[k];
        a.y  = kyrow[k + 1] * u_lds[k + 1];
        b0.x = kxc0[k];
        b0.y = kxc0[k + 1];
        b1.x = kxc1[k];
        b1.y = kxc1[k + 1];
        acc0 = wmma4(a, b0, acc0);                             // 2 independent
        acc1 = wmma4(a, b1, acc1);                             // WMMA chains
      }
      const int yy0 = ty * 16 + 8 * hi;                        // D: M = j + 8*hi
      const int xxA = tx0 * 16 + lm;                           //    N = lm
      const int xxB = (tx0 + 1) * 16 + lm;
#pragma unroll
      for (int j = 0; j < 8; ++j) {
        const int s0 = (yy0 + j) * S_DIM + xxA;
        const int s1 = (yy0 + j) * S_DIM + xxB;
        v_lds[s0] = fast_div(b[s0], acc0[j] + M_EPSF);
        v_lds[s1] = fast_div(b[s1], acc1[j] + M_EPSF);
      }
    }
    __syncthreads();

    // Phase 2: W[n,yy] = sum_xx Kx[n,xx]*v[yy,xx]  (512x128x128 GEMM)
    //          u[n] = a / (sum_yy Ky[n,yy]*W[n,yy] + eps)
    // wave handles n-rows [16*wave, 16*wave+16); all yy-tiles share A.
    float racc[8];
#pragma unroll
    for (int j = 0; j < 8; ++j) racc[j] = 0.0f;
    const float* arow = Kx + (16 * wave + lm) * S_DIM;         // A row m=lm
    for (int t = 0; t < 8; t += 2) {                           // yy tile pair
      v8f acc0 = {0.f, 0.f, 0.f, 0.f, 0.f, 0.f, 0.f, 0.f};
      v8f acc1 = {0.f, 0.f, 0.f, 0.f, 0.f, 0.f, 0.f, 0.f};
      const float* brow0 = v_lds + (t * 16 + lm) * S_DIM;
      const float* brow1 = v_lds + ((t + 1) * 16 + lm) * S_DIM;
#pragma unroll 4
      for (int k0 = 0; k0 < S_DIM; k0 += 4) {
        const int k = k0 + 2 * hi;
        v2f a, b0, b1;
        a.x  = arow[k];
        a.y  = arow[k + 1];
        b0.x = brow0[k];
        b0.y = brow0[k + 1];
        b1.x = brow1[k];
        b1.y = brow1[k + 1];
        acc0 = wmma4(a, b0, acc0);
        acc1 = wmma4(a, b1, acc1);
      }
      // fuse Ky reduction: element (n = 16w + j + 8*hi, yy = t*16 + lm)
      const float* kyp0 = KyT + (t * 16 + lm) * NPTS + 16 * wave + 8 * hi;
      const float* kyp1 = KyT + ((t + 1) * 16 + lm) * NPTS + 16 * wave + 8 * hi;
#pragma unroll
      for (int j = 0; j < 8; ++j)
        racc[j] += acc0[j] * kyp0[j] + acc1[j] * kyp1[j];
    }
    // sum over the 16 lanes of each half (N-dimension of D)
#pragma unroll
    for (int j = 0; j < 8; ++j) {
      float r = racc[j];
      r += __shfl_xor(r, 1, 32);
      r += __shfl_xor(r, 2, 32);
      r += __shfl_xor(r, 4, 32);
      r += __shfl_xor(r, 8, 32);
      racc[j] = r;
    }
    if (lm == 0) {
#pragma unroll
      for (int j = 0; j < 8; ++j)
        u_lds[16 * wave + 8 * hi + j] =
            fast_div(1.0f / (float)NPTS, racc[j] + M_EPSF);
    }
    __syncthreads();
  }

  // ---------------- epilogue: beta-based scalars ----------------
  float p_sc = 0.f, p_sb = 0.f, p_ot = 0.f;
  for (int s = tid; s < SS; s += THREADS) {
    float beta = REG_F * logf(v_lds[s] + M_EPSF);
    float sdv = sd[s];
    p_sc += sdv;
    p_sb += sdv * beta;
    p_ot += b[s] * beta;
  }
  float sc = blockReduceSum(p_sc, red);
  float sb = blockReduceSum(p_sb, red);
  float ot = blockReduceSum(p_ot, red);
  const float denom = sc * sc + 1e-8f;
  const float g0 = sc / denom;
  const float g1 = sb / denom;

  float p_loss = 0.f;
  for (int s = tid; s < SS; s += THREADS) {
    float beta = REG_F * logf(v_lds[s] + M_EPSF);
    p_loss += sd[s] * (g0 * beta - g1);
  }
  float loss = blockReduceSum(p_loss, red);

  // wd = sum_n u[n] * sum_yy Ky*(dy*sum_xx Kx*v + sum_xx Kx*dx*v)
  // Each point n is split across 2 threads (64 yy rows each).
  float p_wd;
  {
    const int n = tid >> 1;
    const int yh = tid & 1;
    const float x = P[n * 2 + 0], y = P[n * 2 + 1];
    const float un = u_lds[n];
    const float* kxr = Kx + n * S_DIM;
    float acc = 0.f;
    for (int yy = yh * 64; yy < yh * 64 + 64; ++yy) {
      float ky = KyT[yy * NPTS + n];
      float cy = 4.0f * (float)yy + 2.0f;
      float dy = -2.0f * y * cy + y * y + cy * cy;
      const float* vr = v_lds + yy * S_DIM;
      float s1 = 0.f, s2 = 0.f;
      for (int xx = 0; xx < S_DIM; ++xx) {
        float cx = 4.0f * (float)xx + 2.0f;
        float dx = -2.0f * x * cx + x * x + cx * cx;
        float kv = kxr[xx] * vr[xx];
        s1 += kv;
        s2 += kv * dx;
      }
      acc += ky * (dy * s1 + s2);
    }
    p_wd = un * acc;
  }
  float wd = blockReduceSum(p_wd, red);

  if (tid == 0) {
    float* part = ws + PART_OFF + img * 4;
    part[0] = loss;
    part[1] = wd;
    part[2] = ot;
  }
}

__global__ void ot_finalize_kernel(const float* __restrict__ ws,
                                   float* __restrict__ out) {
  const int k = threadIdx.x;
  if (k < 3) {
    float s = 0.f;
    for (int i = 0; i < NIMG; ++i) s += ws[PART_OFF + i * 4 + k];  // fixed order
    out[k] = s;
  }
}

extern "C" void kernel_launch(void* const* d_in, const int* in_sizes, int n_in,
                              void* d_out, int out_size, void* d_ws,
                              size_t ws_size, hipStream_t stream) {
  (void)in_sizes; (void)n_in; (void)out_size; (void)ws_size;
  const float* nd  = (const float*)d_in[0];  // normed_density
  const float* ud  = (const float*)d_in[1];  // unnormed_density
  const float* pts = (const float*)d_in[2];  // points
  // d_in[3] (gt_discrete) is unused by the reference.
  float* ws  = (float*)d_ws;
  float* out = (float*)d_out;

  ot_sinkhorn_kernel<<<NIMG, THREADS, 0, stream>>>(nd, ud, pts, ws);
  ot_finalize_kernel<<<1, 32, 0, stream>>>(ws, out);
}